// DWT_d4_20315195310495
// MI455X (gfx1250) — compile-verified
//
#include <hip/hip_runtime.h>

// ---------------------------------------------------------------------------
// Fused 2-D coif1 DWT (single level, pywt 'symmetric'), f32, CDNA5 (gfx1250).
//   in : (256 images) x 512 x 512 f32
//   out: (256 images) x 4 subbands x 258 x 258 f32, channel = 4*img + {A,H,V,D}
// One wave32 per 16x16 output tile. Strided 6-tap conv cast as banded 36x16
// matrix products on V_WMMA_F32_16X16X4_F32. Interior tiles fetch their
// 36x36 input patch with the Tensor Data Mover (tensor_load_to_lds +
// s_wait_tensorcnt); border tiles use a scalar reflect loop. Banded filter
// fragments come from a compile-time constant table instead of cndmask chains.
// ---------------------------------------------------------------------------

typedef __attribute__((ext_vector_type(2))) float        v2f;
typedef __attribute__((ext_vector_type(8))) float        v8f;
typedef __attribute__((ext_vector_type(4))) unsigned int u32x4;
typedef __attribute__((ext_vector_type(8))) int          i32x8;
typedef __attribute__((ext_vector_type(4))) int          i32x4;

#define IN_H 512
#define IN_W 512
#define OUT_N 258            // (512 + 6 - 1) / 2
#define TILES 17             // ceil(258 / 16)
#define PATCH 36             // 2*16 + (6-2) halo
#define XS_STRIDE 36         // contiguous for TDM

#if __has_builtin(__builtin_amdgcn_tensor_load_to_lds)
#define HAS_TDM 1
#else
#define HAS_TDM 0
#endif

// coif1 decomposition filters, pre-flipped (correlation form == pywt convention)
#define L0f (-0.0727326195128539f)
#define L1f ( 0.3378976624578092f)
#define L2f ( 0.8525720202122554f)
#define L3f ( 0.38486484686420286f)
#define L4f (-0.0727326195128539f)
#define L5f (-0.01565572813546454f)

#define H0f (-0.01565572813546454f)
#define H1f ( 0.0727326195128539f)
#define H2f ( 0.38486484686420286f)
#define H3f (-0.8525720202122554f)
#define H4f ( 0.3378976624578092f)
#define H5f ( 0.0727326195128539f)

// -------- compile-time per-lane WMMA filter-fragment table ------------------
// Banded matrix F[i,k] = f[i - 2k] (0 <= i-2k < 6). Per lane: n = lane&15,
// K-pair base = 4t + 2*(lane>>4). Same values serve stage-1 B fragments and
// stage-2 A fragments. Layout: v[((s*9 + t)*32 + lane)*2 + p], s=0 lo / 1 hi.
struct alignas(16) FiltTab { float v[2 * 9 * 32 * 2]; };

constexpr float bandc(const float* f, int d) {
    return (d >= 0 && d < 6) ? f[d] : 0.0f;
}

constexpr FiltTab make_tab() {
    FiltTab t{};
    const float lo[6] = {L0f, L1f, L2f, L3f, L4f, L5f};
    const float hi[6] = {H0f, H1f, H2f, H3f, H4f, H5f};
    for (int s = 0; s < 2; ++s)
        for (int tt = 0; tt < 9; ++tt)
            for (int lane = 0; lane < 32; ++lane)
                for (int p = 0; p < 2; ++p) {
                    const int n = lane & 15;
                    const int k = 4 * tt + 2 * (lane >> 4) + p;
                    t.v[((s * 9 + tt) * 32 + lane) * 2 + p] =
                        bandc(s ? hi : lo, k - 2 * n);
                }
    return t;
}

__constant__ FiltTab FTAB = make_tab();

// D(16x16,f32) = A(16x4,f32) * B(4x16,f32) + C   -- gfx1250 VOP3P WMMA
__device__ __forceinline__ v8f wmma4(v2f a, v2f b, v8f c) {
    return __builtin_amdgcn_wmma_f32_16x16x4_f32(
        false, a, false, b, (short)0, c, false, false);
}

__global__ __launch_bounds__(32)
void dwt2_coif1_wmma(const float* __restrict__ in, float* __restrict__ out) {
    // Per-wave LDS (single-wave workgroup: no cross-wave sharing, no barriers)
    __shared__ __align__(16) float xs[48 * XS_STRIDE]; // 36x36 patch (+ slack rows)
    __shared__ __align__(16) float tlo[48 * 16];       // stage-1 low-pass  (K-major)
    __shared__ __align__(16) float thi[48 * 16];       // stage-1 high-pass

    const int lane = threadIdx.x;          // wave32
    const int n    = lane & 15;            // N column (B/C/D) and M row (A)
    const int half = lane >> 4;
    const int klo  = half * 2;

    const int wg   = blockIdx.x;
    const int img  = wg / (TILES * TILES);
    const int tile = wg - img * (TILES * TILES);
    const int tr   = tile / TILES;
    const int tc   = tile - tr * TILES;
    const int r0   = tr * 16;
    const int c0   = tc * 16;

    // ---- per-lane banded-filter fragments (constant table, L2-hot) ---------
    v2f fLo[9], fHi[9];
#pragma unroll
    for (int t = 0; t < 9; ++t) {
        fLo[t] = *(const v2f*)&FTAB.v[((0 * 9 + t) * 32 + lane) * 2];
        fHi[t] = *(const v2f*)&FTAB.v[((1 * 9 + t) * 32 + lane) * 2];
    }

    const float* __restrict__ src = in + (size_t)img * (IN_H * IN_W);

    // ---- fetch 36x36 input patch into LDS ----------------------------------
    // output col k uses padded cols 2k..2k+5  ->  input cols 2k-4 .. 2k+1
    const bool interior = (tr >= 1) & (tr <= 15) & (tc >= 1) & (tc <= 15);
#if HAS_TDM
    if (interior) {
        // Tensor Data Mover: 2-D tile (36x36 f32) of a 512-stride tensor.
        const unsigned lds_base = (unsigned)(size_t)&xs[0]; // flat LDS addr[31:0] == offset
        const unsigned long long ga =
            (unsigned long long)(size_t)(src + (size_t)(2 * r0 - 4) * IN_W + (2 * c0 - 4));
        // D# group0: count=1 | lds_addr | global_addr[56:0] | type=2
        u32x4 g0 = { 1u, lds_base, (unsigned)ga,
                     (unsigned)((ga >> 32) & 0x1FFFFFFu) | (2u << 30) };
        // D# group1: data_size=4B; tensor_dim0/1=512; tile_dim0/1=36; dim0_stride=512
        i32x8 g1 = { 0x00020000,        // data_size=2 (4 bytes), no pad/multicast
                     512 << 16,         // tensor_dim0 lo16 (hi16 -> g1[2][15:0]=0)
                     512 << 16,         // tensor_dim1 lo16
                     36 << 16,          // tile_dim0
                     36,                // tile_dim1 (tile_dim2=0)
                     512,               // tensor_dim0_stride lo32
                     0, 0 };            // dim0_stride hi16, dim1_stride = 0 (2-D)
        i32x4 g2 = { 0, 0, 0, 0 };
        i32x4 g3 = { 0, 0, 0, 0 };
        i32x8 g4 = { 0, 0, 0, 0, 0, 0, 0, 0 };   // extra group (clang-23 form): zero
        __builtin_amdgcn_tensor_load_to_lds(g0, g1, g2, g3, g4, 0);
        __builtin_amdgcn_s_wait_tensorcnt(0);
    } else
#endif
    {
        for (int idx = lane; idx < PATCH * PATCH; idx += 32) {
            const int i = idx / PATCH;
            const int j = idx - i * PATCH;
            int gi = 2 * r0 - 4 + i;
            gi = (gi < 0) ? (-1 - gi) : ((gi >= IN_H) ? (2 * IN_H - 1 - gi) : gi);
            int gj = 2 * c0 - 4 + j;
            gj = (gj < 0) ? (-1 - gj) : ((gj >= IN_W) ? (2 * IN_W - 1 - gj) : gj);
            xs[i * XS_STRIDE + j] = src[gi * IN_W + gj];
        }
    }
    __builtin_amdgcn_wave_barrier();       // keep compiler from hoisting DS reads

    // ---- stage 1: along W.  T(36x16) = X(36x36) * F(36x16) -----------------
    v8f accLo[3], accHi[3];
#pragma unroll
    for (int m = 0; m < 3; ++m) { accLo[m] = (v8f){}; accHi[m] = (v8f){}; }

#pragma unroll
    for (int t = 0; t < 9; ++t) {          // K = 36 -> 9 steps of 4
        const int k0 = 4 * t + klo;
#pragma unroll
        for (int m = 0; m < 3; ++m) {      // 36 patch rows -> 3 M-blocks of 16
            // A: M = n, K = k0,k0+1  (8B-aligned ds_load_b64)
            v2f a = *(const v2f*)&xs[(16 * m + n) * XS_STRIDE + k0];
            accLo[m] = wmma4(a, fLo[t], accLo[m]);
            accHi[m] = wmma4(a, fHi[t], accHi[m]);
        }
    }

    // park stage-1 results K-major: t{lo,hi}[input_row][out_col]
#pragma unroll
    for (int m = 0; m < 3; ++m)
#pragma unroll
        for (int v = 0; v < 8; ++v) {
            const int rr = 16 * m + v + 8 * half;           // C/D layout row
            tlo[rr * 16 + n] = accLo[m][v];
            thi[rr * 16 + n] = accHi[m][v];
        }
    __builtin_amdgcn_wave_barrier();

    // ---- stage 2: along H.  S(16x16) = G(16x36) * T(36x16) -----------------
    // G has the same banded structure; per-lane A-fragment values == fLo/fHi.
    v8f acc[4];
#pragma unroll
    for (int s = 0; s < 4; ++s) acc[s] = (v8f){};

#pragma unroll
    for (int t = 0; t < 9; ++t) {
        const int k0 = 4 * t + klo;
        v2f bl, bh;                         // B fragment: rows k0,k0+1, col n
        bl.x = tlo[k0 * 16 + n];  bl.y = tlo[(k0 + 1) * 16 + n];
        bh.x = thi[k0 * 16 + n];  bh.y = thi[(k0 + 1) * 16 + n];
        acc[0] = wmma4(fLo[t], bl, acc[0]); // cA = lo(H) . lo(W)
        acc[1] = wmma4(fHi[t], bl, acc[1]); // cH = hi(H) . lo(W)
        acc[2] = wmma4(fLo[t], bh, acc[2]); // cV = lo(H) . hi(W)
        acc[3] = wmma4(fHi[t], bh, acc[3]); // cD = hi(H) . hi(W)
    }

    // ---- store: out[(4*img + s)][r0+row][c0+col] ----------------------------
    const size_t plane = (size_t)OUT_N * OUT_N;
    float* __restrict__ o0 = out + (size_t)(4 * img) * plane;
    const int col = c0 + n;
    if ((tr < 16) & (tc < 16)) {            // uniform fast path: no edge guards
#pragma unroll
        for (int s = 0; s < 4; ++s)
#pragma unroll
            for (int v = 0; v < 8; ++v) {
                const int row = r0 + v + 8 * half;
                o0[(size_t)s * plane + (size_t)row * OUT_N + col] = acc[s][v];
            }
    } else {                                // ragged 258 edge
#pragma unroll
        for (int s = 0; s < 4; ++s)
#pragma unroll
            for (int v = 0; v < 8; ++v) {
                const int row = r0 + v + 8 * half;
                if (row < OUT_N && col < OUT_N)
                    o0[(size_t)s * plane + (size_t)row * OUT_N + col] = acc[s][v];
            }
    }
}

extern "C" void kernel_launch(void* const* d_in, const int* in_sizes, int n_in,
                              void* d_out, int out_size, void* d_ws, size_t ws_size,
                              hipStream_t stream) {
    (void)n_in; (void)out_size; (void)d_ws; (void)ws_size;
    const float* x = (const float*)d_in[0];
    float* out = (float*)d_out;
    const int n_img = in_sizes[0] / (IN_H * IN_W);          // B*C = 256
    dim3 grid((unsigned)(n_img * TILES * TILES));           // 256 * 289 waves
    dim3 block(32);                                         // one wave32 per tile
    dwt2_coif1_wmma<<<grid, block, 0, stream>>>(x, out);
}